// Windowed_Matching_Loss_28587302323005
// MI455X (gfx1250) — compile-verified
//
#include <hip/hip_runtime.h>
#include <stdint.h>

// Problem constants (match setup_inputs: B=1, H=240, W=320)
#define HH 240
#define WW 320
#define NPIX (HH * WW)
#define EPSV 1e-5f

typedef __attribute__((ext_vector_type(2))) float v2f;
typedef __attribute__((ext_vector_type(8))) float v8f;

// ---------------------------------------------------------------------------
// gfx1250 async global->LDS helpers (CDNA5 path, CDNA5 ISA)
// Per-lane copy of 4 bytes from global memory into LDS, tracked by ASYNCcnt.
// ---------------------------------------------------------------------------
__device__ __forceinline__ void async_g2l_b32(void* lds_ptr, const void* gptr) {
    uint32_t lds_addr = (uint32_t)(uintptr_t)lds_ptr;
    uint64_t gaddr    = (uint64_t)(uintptr_t)gptr;
    asm volatile("global_load_async_to_lds_b32 %0, %1, off"
                 :
                 : "v"(lds_addr), "v"(gaddr)
                 : "memory");
}
__device__ __forceinline__ void wait_async0() {
    asm volatile("s_wait_asynccnt 0" ::: "memory");
}

// ---------------------------------------------------------------------------
// Kernel 1: LCN (9x9 zero-padded box stats) via banded-ones WMMA.
// One wave32 per 16x16 output tile; halo 4 -> 24x24 region (padded to 24x32).
//   vertical:   V = BandV(16x24) * X(24x16)   (6 chained K=4 WMMAs per group)
//   horizontal: S = V(16x24) * BandH(24x16)   (6 chained K=4 WMMAs)
// Sum-of-squares shares the band operand (B fragment squared elementwise).
// Exact in f32 (ones * value, f32 accumulate).
// ---------------------------------------------------------------------------
#define LT 16
#define LRH2 24      // region height (16 + 2*4)
#define LRWP 32      // region width padded (24 valid, zero beyond)

__global__ __launch_bounds__(32) void lcn_wmma_kernel(const float* __restrict__ img,
                                                      float* __restrict__ lcn,
                                                      float* __restrict__ stdo) {
    __shared__ float sX[LRH2][LRWP]; // input region, zero padded
    __shared__ float sV[16][LRWP];   // vertical box sums
    __shared__ float sQ[16][LRWP];   // vertical box sums of squares
    const int l  = threadIdx.x;           // 0..31 (wave32)
    const int hi = (l >= 16) ? 1 : 0;
    const int lm = l & 15;
    const int x0 = blockIdx.x * LT, y0 = blockIdx.y * LT;

    // zero-fill region (zero padding for OOB taps + padded columns)
    for (int i = l; i < LRH2 * LRWP; i += 32) (&sX[0][0])[i] = 0.0f;
    __syncthreads();
    // async fill of in-bounds 24x24 halo region
    for (int i = l; i < LRH2 * 24; i += 32) {
        int ry = i / 24, rx = i - ry * 24;
        int gx = x0 - 4 + rx, gy = y0 - 4 + ry;
        if (gx >= 0 && gx < WW && gy >= 0 && gy < HH)
            async_g2l_b32(&sX[ry][rx], img + gy * WW + gx);
    }
    wait_async0();
    __syncthreads();

    // ---- vertical pass: two 16-column groups ----
    for (int g = 0; g < 2; ++g) {
        v8f dS = {}; v8f dQ = {};
        for (int c = 0; c < 6; ++c) {
            v2f a, bS, bQ;
#pragma unroll
            for (int v = 0; v < 2; ++v) {
                int k = 4 * c + v + 2 * hi; // K index of this slot (A and B share split)
                // A (banded ones), documented layout: M = l&15, K = k
                a[v] = (k >= lm && k <= lm + 8) ? 1.0f : 0.0f;
                // B (data), K-split mirrored layout: row K = k, col N = l&15
                float xv = sX[k][g * 16 + lm];
                bS[v] = xv;
                bQ[v] = xv * xv;
            }
            dS = __builtin_amdgcn_wmma_f32_16x16x4_f32(false, a, false, bS, (short)0, dS, false, false);
            dQ = __builtin_amdgcn_wmma_f32_16x16x4_f32(false, a, false, bQ, (short)0, dQ, false, false);
        }
        // D layout (documented): VGPR v -> row M = v + 8*(l>=16), col N = l&15
#pragma unroll
        for (int v = 0; v < 8; ++v) {
            int m = v + 8 * hi;
            sV[m][g * 16 + lm] = dS[v];
            sQ[m][g * 16 + lm] = dQ[v];
        }
    }
    __syncthreads();

    // ---- horizontal pass: S = V * BandH ----
    v8f dS2 = {}; v8f dQ2 = {};
    for (int c = 0; c < 6; ++c) {
        v2f aS, aQ, b;
#pragma unroll
        for (int v = 0; v < 2; ++v) {
            int k = 4 * c + v + 2 * hi;
            // A (data), documented layout: row M = l&15, K = k
            aS[v] = sV[lm][k];
            aQ[v] = sQ[lm][k];
            // B (banded ones): B[k][n] = 1 iff n <= k <= n+8, n = l&15
            b[v] = (k >= lm && k <= lm + 8) ? 1.0f : 0.0f;
        }
        dS2 = __builtin_amdgcn_wmma_f32_16x16x4_f32(false, aS, false, b, (short)0, dS2, false, false);
        dQ2 = __builtin_amdgcn_wmma_f32_16x16x4_f32(false, aQ, false, b, (short)0, dQ2, false, false);
    }

    // ---- finalize LCN per held D element ----
    const float inv_n = 1.0f / 81.0f;
#pragma unroll
    for (int v = 0; v < 8; ++v) {
        int m = v + 8 * hi, n = lm;
        float s = dS2[v], q = dQ2[v];
        float avg = s * inv_n;
        float var = fmaxf(q * inv_n - avg * avg, 0.0f);
        float sd  = sqrtf(var);
        float center = sX[m + 4][n + 4];
        int gidx = (y0 + m) * WW + (x0 + n);
        lcn[gidx]  = (center - avg) / (sd + EPSV);
        stdo[gidx] = sd;
    }
}

// ---------------------------------------------------------------------------
// Kernel 2: disparity warp of lcn_R (grid_sample, zeros padding) and
// per-pixel cost C = std_L * |lcn_L - warp(lcn_R)|.
// ---------------------------------------------------------------------------
__global__ __launch_bounds__(256) void warp_cost_kernel(const float* __restrict__ lcnL,
                                                        const float* __restrict__ stdL,
                                                        const float* __restrict__ lcnR,
                                                        const float* __restrict__ disp,
                                                        float* __restrict__ Cb) {
    int i = blockIdx.x * 256 + threadIdx.x;
    if (i >= NPIX) return;
    int y = i / WW, x = i - y * WW;
    float fx = (float)x - disp[i];
    float fy = (float)y;
    // normalize (align_corners=True convention) then unnormalize (align_corners=False)
    float xn = (fx - (float)(WW - 1) * 0.5f) / (float)(WW - 1) * 2.0f;
    float yn = (fy - (float)(HH - 1) * 0.5f) / (float)(HH - 1) * 2.0f;
    float ix = ((xn + 1.0f) * (float)WW - 1.0f) * 0.5f;
    float iy = ((yn + 1.0f) * (float)HH - 1.0f) * 0.5f;
    float x0f = floorf(ix), y0f = floorf(iy);
    float x1f = x0f + 1.0f, y1f = y0f + 1.0f;

    auto gather = [&](float xf, float yf) -> float {
        bool valid = (xf >= 0.0f) && (xf <= (float)(WW - 1)) &&
                     (yf >= 0.0f) && (yf <= (float)(HH - 1));
        int xi = (int)xf, yi = (int)yf;
        xi = xi < 0 ? 0 : (xi > WW - 1 ? WW - 1 : xi);
        yi = yi < 0 ? 0 : (yi > HH - 1 ? HH - 1 : yi);
        float v = lcnR[yi * WW + xi];
        return valid ? v : 0.0f;
    };

    float wa = (x1f - ix) * (y1f - iy);
    float wb = (x1f - ix) * (iy - y0f);
    float wc = (ix - x0f) * (y1f - iy);
    float wd = (ix - x0f) * (iy - y0f);
    float rep = wa * gather(x0f, y0f) + wb * gather(x0f, y1f) +
                wc * gather(x1f, y0f) + wd * gather(x1f, y1f);
    Cb[i] = stdL[i] * fabsf(lcnL[i] - rep);
}

// ---------------------------------------------------------------------------
// Kernel 3: 33x33 bilateral aggregation (dominant cost, ~84M v_exp_f32).
// 16x16 output tile, radius 16 -> 48x48 LDS tiles of lcn_L and C
// (zero-padded), async-filled. Deterministic per-block tree reduction.
// ---------------------------------------------------------------------------
#define ATW 16
#define WRAD 16
#define AR (ATW + 2 * WRAD) // 48

__global__ __launch_bounds__(256) void agg_kernel(const float* __restrict__ lcnL,
                                                  const float* __restrict__ Cb,
                                                  float* __restrict__ partial) {
    __shared__ float sL[AR * AR];
    __shared__ float sC[AR * AR];
    __shared__ float red[256];
    const int tx = threadIdx.x, ty = threadIdx.y;
    const int tid = ty * ATW + tx;
    const int x0 = blockIdx.x * ATW, y0 = blockIdx.y * ATW;

    for (int i = tid; i < AR * AR; i += 256) {
        sL[i] = 0.0f;
        sC[i] = 0.0f;
    }
    __syncthreads();
    for (int i = tid; i < AR * AR; i += 256) {
        int ry = i / AR, rx = i - ry * AR;
        int gx = x0 - WRAD + rx, gy = y0 - WRAD + ry;
        if (gx >= 0 && gx < WW && gy >= 0 && gy < HH) {
            int g = gy * WW + gx;
            async_g2l_b32(&sL[i], lcnL + g);
            async_g2l_b32(&sC[i], Cb + g);
        }
    }
    wait_async0();
    __syncthreads();

    float c0 = sL[(ty + WRAD) * AR + (tx + WRAD)];
    float num = 0.0f, den = 0.0f;
#pragma unroll 1
    for (int dy = 0; dy < 33; ++dy) {
        const float* rowL = &sL[(ty + dy) * AR + tx];
        const float* rowC = &sC[(ty + dy) * AR + tx];
#pragma unroll 3
        for (int dx = 0; dx < 33; ++dx) {
            float w = __expf(-0.5f * fabsf(c0 - rowL[dx]));
            num = fmaf(rowC[dx], w, num);
            den += w;
        }
    }
    float cw = num / den;

    __syncthreads();
    red[tid] = cw;
    __syncthreads();
#pragma unroll
    for (int s2 = 128; s2 > 0; s2 >>= 1) {
        if (tid < s2) red[tid] += red[tid + s2];
        __syncthreads();
    }
    if (tid == 0) partial[blockIdx.y * gridDim.x + blockIdx.x] = red[0];
}

// ---------------------------------------------------------------------------
// Kernel 4: deterministic final reduction (fixed order, graph-replay stable).
// ---------------------------------------------------------------------------
__global__ void finalize_kernel(const float* __restrict__ partial, int nblk,
                                float* __restrict__ out) {
    if (threadIdx.x == 0 && blockIdx.x == 0) {
        float s = 0.0f;
        for (int i = 0; i < nblk; ++i) s += partial[i];
        out[0] = s / (float)NPIX;
    }
}

// ---------------------------------------------------------------------------
extern "C" void kernel_launch(void* const* d_in, const int* in_sizes, int n_in,
                              void* d_out, int out_size, void* d_ws, size_t ws_size,
                              hipStream_t stream) {
    const float* imgL = (const float*)d_in[0];
    const float* imgR = (const float*)d_in[1];
    const float* disp = (const float*)d_in[2];

    float* ws      = (float*)d_ws;
    float* lcnL    = ws;
    float* stdL    = ws + (size_t)NPIX;
    float* lcnR    = ws + (size_t)2 * NPIX;
    float* stdR    = ws + (size_t)3 * NPIX;
    float* Cb      = ws + (size_t)4 * NPIX;
    float* partial = ws + (size_t)5 * NPIX; // 300 floats

    dim3 lb(32, 1);
    dim3 lg(WW / LT, HH / LT); // 20 x 15, one wave per 16x16 tile
    hipLaunchKernelGGL(lcn_wmma_kernel, lg, lb, 0, stream, imgL, lcnL, stdL);
    hipLaunchKernelGGL(lcn_wmma_kernel, lg, lb, 0, stream, imgR, lcnR, stdR);

    hipLaunchKernelGGL(warp_cost_kernel, dim3((NPIX + 255) / 256), dim3(256), 0,
                       stream, lcnL, stdL, lcnR, disp, Cb);

    dim3 ab(ATW, ATW);
    dim3 ag(WW / ATW, HH / ATW); // 20 x 15 = 300 blocks
    hipLaunchKernelGGL(agg_kernel, ag, ab, 0, stream, lcnL, Cb, partial);

    hipLaunchKernelGGL(finalize_kernel, dim3(1), dim3(32), 0, stream, partial,
                       (int)(ag.x * ag.y), (float*)d_out);
}